// VIT_85779086836196
// MI455X (gfx1250) — compile-verified
//
#include <hip/hip_runtime.h>
#include <hip/hip_bf16.h>

// ---------------------------------------------------------------------------
// ViT-Base forward (B=32, S=197, D=768, L=12, NH=12, DH=64, MD=3072, OUT=1000)
// fp32 throughout, matmuls on V_WMMA_F32_16X16X4_F32 (wave32).
// Each wave computes a 64x64 C tile (4x4 WMMA subtiles, 16 v8f accumulators).
// Ragged M/N handled by clamping A/B row pointers (bad rows/cols are computed
// but never stored), so the K-loop has NO branches and NO selects.
// ---------------------------------------------------------------------------

#define VB    32
#define VC    3
#define VHW   224
#define VNP   14
#define VPS   16
#define VS    197      // tokens
#define VD    768
#define VNH   12
#define VDH   64
#define VMD   3072
#define VOUT  1000
#define VL    12
#define NPAT  196      // 14*14

typedef float v2f __attribute__((ext_vector_type(2)));
typedef float v8f __attribute__((ext_vector_type(8)));

#define GF_GELU   1
#define GF_ADDC   2
#define GF_TRANSC 4

// ---------------------------------------------------------------------------
// Generic batched GEMM:  C[m,n] = act(alpha * sum_k A[m,k]*B[n,k] + bias[n]) (+= C)
// Batch index z = i1*n2 + i2 applies separate pointer strides per dimension.
//
// WMMA f32 16x16x4 lane layout (ISA 7.12.2):
//   A (16x4): lane L holds A[L%16][k0 + 2*(L/16) + {0,1}] in 2 VGPRs
//   B (4x16): lane L holds B[k0 + 2*(L/16) + {0,1}][L%16]  (B[n,k] here = W[n][k])
//   C (16x16): lane L, vgpr r -> row (r + 8*(L/16)), col L%16
// ---------------------------------------------------------------------------
__global__ __launch_bounds__(256)
void gemm_wmma_f32(const float* __restrict__ A, int lda, long long sA1, long long sA2,
                   const float* __restrict__ Bw, int ldb, long long sB1, long long sB2,
                   const float* __restrict__ bias, long long sBi1, long long sBi2,
                   float* __restrict__ C, int ldc, long long sC1, long long sC2,
                   int M, int N, int K, int n2, float alpha, int flags)
{
    const int z  = blockIdx.z;
    const int i1 = z / n2;
    const int i2 = z % n2;
    A  += (long long)i1 * sA1 + (long long)i2 * sA2;
    Bw += (long long)i1 * sB1 + (long long)i2 * sB2;
    if (bias) bias += (long long)i1 * sBi1 + (long long)i2 * sBi2;
    C  += (long long)i1 * sC1 + (long long)i2 * sC2;

    const int tm = (M + 63) >> 6;
    const int tn = (N + 63) >> 6;
    const int wave = threadIdx.x >> 5;
    const int t = blockIdx.x * (blockDim.x >> 5) + wave;
    if (t >= tm * tn) return;              // wave-uniform exit: EXEC stays all-ones
    const int tM = (t / tn) << 6;
    const int tN = (t % tn) << 6;

    const int lane = threadIdx.x & 31;
    const int idx  = lane & 15;            // m (for A) / n (for B,C) within subtile
    const int kh   = (lane >> 4) << 1;     // 0 or 2: K sub-offset for this half-wave

    // Row/col pointers, clamped into range: out-of-range rows/cols produce
    // garbage in output rows/cols that the epilogue never stores.
    const float* aR[4];
    const float* bR[4];
#pragma unroll
    for (int i = 0; i < 4; ++i) {
        int m = tM + i * 16 + idx; if (m > M - 1) m = M - 1;
        int n = tN + i * 16 + idx; if (n > N - 1) n = N - 1;
        aR[i] = A  + (long long)m * lda;
        bR[i] = Bw + (long long)n * ldb;
    }

    v8f acc[16];
#pragma unroll
    for (int i = 0; i < 16; ++i)
        acc[i] = (v8f){0.f, 0.f, 0.f, 0.f, 0.f, 0.f, 0.f, 0.f};

    const int Kf = K & ~3;
    for (int k0 = 0; k0 < Kf; k0 += 4) {
        const int kk = k0 + kh;
        v2f a[4], b[4];
#pragma unroll
        for (int i = 0; i < 4; ++i) {
            a[i].x = aR[i][kk]; a[i].y = aR[i][kk + 1];
            b[i].x = bR[i][kk]; b[i].y = bR[i][kk + 1];
        }
#pragma unroll
        for (int i = 0; i < 4; ++i)
#pragma unroll
            for (int j = 0; j < 4; ++j)
                acc[i * 4 + j] = __builtin_amdgcn_wmma_f32_16x16x4_f32(
                    false, a[i], false, b[j], (short)0, acc[i * 4 + j], false, false);
    }

    if (Kf < K) {   // ragged-K tail (K=197 att@V): clamped loads + branch-free selects
        const int kk = Kf + kh;
        const int k1 = (kk     < K) ? kk     : K - 1;
        const int k2 = (kk + 1 < K) ? kk + 1 : K - 1;
        const bool v1 = kk < K, v2 = (kk + 1) < K;
        v2f a[4], b[4];
#pragma unroll
        for (int i = 0; i < 4; ++i) {
            float ax = aR[i][k1], ay = aR[i][k2];
            float bx = bR[i][k1], by = bR[i][k2];
            a[i].x = v1 ? ax : 0.f;  a[i].y = v2 ? ay : 0.f;
            b[i].x = v1 ? bx : 0.f;  b[i].y = v2 ? by : 0.f;
        }
#pragma unroll
        for (int i = 0; i < 4; ++i)
#pragma unroll
            for (int j = 0; j < 4; ++j)
                acc[i * 4 + j] = __builtin_amdgcn_wmma_f32_16x16x4_f32(
                    false, a[i], false, b[j], (short)0, acc[i * 4 + j], false, false);
    }

    // ---- epilogue (divergence allowed here; no more WMMA) ----
    const int roff = (lane >> 4) << 3;     // 0 or 8
#pragma unroll
    for (int j = 0; j < 4; ++j) {
        const int n = tN + j * 16 + idx;
        if (n >= N) continue;
        const float bval = bias ? bias[n] : 0.f;
#pragma unroll
        for (int i = 0; i < 4; ++i) {
            const int mb = tM + i * 16 + roff;
#pragma unroll
            for (int r = 0; r < 8; ++r) {
                const int m = mb + r;
                if (m < M) {
                    float v = acc[i * 4 + j][r] * alpha + bval;
                    if (flags & GF_GELU)
                        v = 0.5f * v * (1.f + erff(v * 0.70710678118654752f));
                    float* p = (flags & GF_TRANSC) ? (C + (long long)n * ldc + m)
                                                   : (C + (long long)m * ldc + n);
                    if (flags & GF_ADDC) v += *p;
                    *p = v;
                }
            }
        }
    }
}

// ---------------------------------------------------------------------------
// patches[b, pr*14+pc, c*256 + i*16 + j] = images[b, c, pr*16+i, pc*16+j]
// ---------------------------------------------------------------------------
__global__ void patchify_kernel(const float* __restrict__ img,
                                float* __restrict__ patches, int total)
{
    int i = blockIdx.x * blockDim.x + threadIdx.x;
    if (i >= total) return;
    int d = i % VD;
    int p = (i / VD) % NPAT;
    int b = i / (VD * NPAT);
    int c   = d >> 8;          // / 256
    int rem = d & 255;
    int pi  = rem >> 4, pj = rem & 15;
    int pr  = p / VNP,  pc = p % VNP;
    int row = pr * VPS + pi, col = pc * VPS + pj;
    patches[i] = img[(((size_t)b * VC + c) * VHW + row) * VHW + col];
}

// x[b,0,:] = cls + pos[0];  x[b,s,:] = tokens[b*196+s-1] + pos[s]
__global__ void assemble_x_kernel(const float* __restrict__ tokens,
                                  const float* __restrict__ cls,
                                  const float* __restrict__ pos,
                                  float* __restrict__ x, int total)
{
    int i = blockIdx.x * blockDim.x + threadIdx.x;
    if (i >= total) return;
    int d = i % VD;
    int s = (i / VD) % VS;
    int b = i / (VD * VS);
    float v = (s == 0) ? cls[d]
                       : tokens[((size_t)b * NPAT + (s - 1)) * VD + d];
    x[i] = v + pos[(size_t)s * VD + d];
}

// ---------------------------------------------------------------------------
// LayerNorm: one 256-thread block per row of length D=768
// ---------------------------------------------------------------------------
__global__ __launch_bounds__(256)
void layernorm_kernel(const float* __restrict__ x, const float* __restrict__ g,
                      const float* __restrict__ b, float* __restrict__ out, int D)
{
    const int row = blockIdx.x;
    const float* xr = x + (size_t)row * D;
    float* orow = out + (size_t)row * D;
    __shared__ float s1[256], s2[256];
    float lsum = 0.f, lsq = 0.f;
    for (int i = threadIdx.x; i < D; i += 256) {
        float v = xr[i]; lsum += v; lsq += v * v;
    }
    s1[threadIdx.x] = lsum; s2[threadIdx.x] = lsq;
    __syncthreads();
    for (int s = 128; s > 0; s >>= 1) {
        if (threadIdx.x < s) {
            s1[threadIdx.x] += s1[threadIdx.x + s];
            s2[threadIdx.x] += s2[threadIdx.x + s];
        }
        __syncthreads();
    }
    const float mean = s1[0] / D;
    const float var  = s2[0] / D - mean * mean;
    const float inv  = rsqrtf(var + 1e-5f);
    for (int i = threadIdx.x; i < D; i += 256)
        orow[i] = (xr[i] - mean) * inv * g[i] + b[i];
}

// ---------------------------------------------------------------------------
// Row softmax: one 256-thread block per row (in-place safe)
// ---------------------------------------------------------------------------
__global__ __launch_bounds__(256)
void softmax_kernel(const float* __restrict__ in, float* __restrict__ out, int len)
{
    const int row = blockIdx.x;
    const float* ir = in + (size_t)row * len;
    float* orow = out + (size_t)row * len;
    __shared__ float sd[256];

    float lmax = -3.402823466e38f;
    for (int i = threadIdx.x; i < len; i += 256) lmax = fmaxf(lmax, ir[i]);
    sd[threadIdx.x] = lmax;
    __syncthreads();
    for (int s = 128; s > 0; s >>= 1) {
        if (threadIdx.x < s) sd[threadIdx.x] = fmaxf(sd[threadIdx.x], sd[threadIdx.x + s]);
        __syncthreads();
    }
    const float mx = sd[0];
    __syncthreads();

    float lsum = 0.f;
    for (int i = threadIdx.x; i < len; i += 256) lsum += expf(ir[i] - mx);
    sd[threadIdx.x] = lsum;
    __syncthreads();
    for (int s = 128; s > 0; s >>= 1) {
        if (threadIdx.x < s) sd[threadIdx.x] += sd[threadIdx.x + s];
        __syncthreads();
    }
    const float inv = 1.f / sd[0];
    for (int i = threadIdx.x; i < len; i += 256)
        orow[i] = expf(ir[i] - mx) * inv;
}

// ---------------------------------------------------------------------------
extern "C" void kernel_launch(void* const* d_in, const int* in_sizes, int n_in,
                              void* d_out, int out_size, void* d_ws, size_t ws_size,
                              hipStream_t stream)
{
    const float* images = (const float*)d_in[0];
    const float* Wp     = (const float*)d_in[1];
    const float* bp     = (const float*)d_in[2];
    const float* cls    = (const float*)d_in[3];
    const float* pos    = (const float*)d_in[4];
    const float* ln1_g  = (const float*)d_in[5];
    const float* ln1_b  = (const float*)d_in[6];
    const float* Wq     = (const float*)d_in[7];
    const float* bq     = (const float*)d_in[8];
    const float* Wk     = (const float*)d_in[9];
    const float* bk     = (const float*)d_in[10];
    const float* Wv     = (const float*)d_in[11];
    const float* bv     = (const float*)d_in[12];
    const float* ln2_g  = (const float*)d_in[13];
    const float* ln2_b  = (const float*)d_in[14];
    const float* W1     = (const float*)d_in[15];
    const float* b1     = (const float*)d_in[16];
    const float* W2     = (const float*)d_in[17];
    const float* b2     = (const float*)d_in[18];
    const float* Wh     = (const float*)d_in[19];
    const float* bh     = (const float*)d_in[20];

    // ---- workspace carve-out (floats) ----
    const size_t nPatch  = (size_t)VB * NPAT * VD;             // 4,816,896
    const size_t nX      = (size_t)VB * VS * VD;               // 4,841,472
    const size_t nHead   = (size_t)VB * VNH * VS * VDH;        // 4,841,472
    const size_t nScores = (size_t)VB * VNH * VS * VS;         // 14,902,656
    const size_t nM1     = (size_t)VB * VS * VMD;              // 19,365,888
    const size_t nLogits = (size_t)VB * VOUT;

    const size_t needed = (nPatch * 2 + nX * 2 + nHead * 3 + nScores + nM1 + nLogits)
                          * sizeof(float);
    if (ws_size < needed) return;

    float* ws      = (float*)d_ws;
    float* patches = ws;  ws += nPatch;
    float* tokens  = ws;  ws += nPatch;
    float* x       = ws;  ws += nX;
    float* hln     = ws;  ws += nX;
    float* q       = ws;  ws += nHead;
    float* kbuf    = ws;  ws += nHead;
    float* vT      = ws;  ws += nHead;     // stored [B,NH,DH,S]
    float* scores  = ws;  ws += nScores;   // [B,NH,S,S]
    float* m1      = ws;  ws += nM1;
    float* logits  = ws;  ws += nLogits;

    auto gemm = [&](const float* A, int lda, long long sA1, long long sA2,
                    const float* Bm, int ldb, long long sB1, long long sB2,
                    const float* bias, long long sBi1, long long sBi2,
                    float* C, int ldc, long long sC1, long long sC2,
                    int M, int N, int K, int n1, int n2, float alpha, int flags)
    {
        int tm = (M + 63) / 64, tn = (N + 63) / 64;
        int tiles = tm * tn;
        dim3 grid((tiles + 7) / 8, 1, n1 * n2);    // 8 waves (256 threads) per block
        gemm_wmma_f32<<<grid, 256, 0, stream>>>(
            A, lda, sA1, sA2, Bm, ldb, sB1, sB2, bias, sBi1, sBi2,
            C, ldc, sC1, sC2, M, N, K, n2, alpha, flags);
    };

    const float attScale = 0.125f;   // 1/sqrt(64)
    const int BS = VB * VS;          // 6304 rows

    // ---- embed ----
    {
        int tot = (int)nPatch;
        patchify_kernel<<<(tot + 255) / 256, 256, 0, stream>>>(images, patches, tot);
        // tokens = patches @ Wp^T + bp     [6272,768] x [768,768]
        gemm(patches, VD, 0, 0, Wp, VD, 0, 0, bp, 0, 0,
             tokens, VD, 0, 0, VB * NPAT, VD, VD, 1, 1, 1.f, 0);
        int totx = (int)nX;
        assemble_x_kernel<<<(totx + 255) / 256, 256, 0, stream>>>(tokens, cls, pos, x, totx);
    }

    // ---- transformer layers ----
    for (int l = 0; l < VL; ++l) {
        const float* wq = Wq + (size_t)l * VNH * VDH * VDH;
        const float* wk = Wk + (size_t)l * VNH * VDH * VDH;
        const float* wv = Wv + (size_t)l * VNH * VDH * VDH;
        const float* bql = bq + (size_t)l * VNH * VDH;
        const float* bkl = bk + (size_t)l * VNH * VDH;
        const float* bvl = bv + (size_t)l * VNH * VDH;

        // LN1
        layernorm_kernel<<<BS, 256, 0, stream>>>(x, ln1_g + (size_t)l * VD,
                                                 ln1_b + (size_t)l * VD, hln, VD);

        // Q,K,V per (b,h):  [S,64] = hln[:, h*64:h*64+64] @ W[h]^T + b[h]
        const long long sAh = (long long)VS * VD;   // per batch
        const long long sHd = (long long)VS * VDH;  // per head in q/k
        gemm(hln, VD, sAh, VDH, wq, VDH, 0, (long long)VDH * VDH, bql, 0, VDH,
             q, VDH, (long long)VNH * VS * VDH, sHd, VS, VDH, VDH, VB, VNH, 1.f, 0);
        gemm(hln, VD, sAh, VDH, wk, VDH, 0, (long long)VDH * VDH, bkl, 0, VDH,
             kbuf, VDH, (long long)VNH * VS * VDH, sHd, VS, VDH, VDH, VB, VNH, 1.f, 0);
        // V stored transposed: vT[b,h,e,s]
        gemm(hln, VD, sAh, VDH, wv, VDH, 0, (long long)VDH * VDH, bvl, 0, VDH,
             vT, VS, (long long)VNH * VDH * VS, (long long)VDH * VS,
             VS, VDH, VDH, VB, VNH, 1.f, GF_TRANSC);

        // scores = (Q @ K^T) * scale     per (b,h): [S,64] x [S,64]^T
        gemm(q, VDH, (long long)VNH * VS * VDH, sHd,
             kbuf, VDH, (long long)VNH * VS * VDH, sHd,
             nullptr, 0, 0,
             scores, VS, (long long)VNH * VS * VS, (long long)VS * VS,
             VS, VS, VDH, VB, VNH, attScale, 0);

        // softmax over last dim (rows = B*NH*S, len = S)
        softmax_kernel<<<VB * VNH * VS, 256, 0, stream>>>(scores, scores, VS);

        // x += att @ V     per (b,h): [S,S] x vT[64,S]^T -> written into x cols h*64..
        gemm(scores, VS, (long long)VNH * VS * VS, (long long)VS * VS,
             vT, VS, (long long)VNH * VDH * VS, (long long)VDH * VS,
             nullptr, 0, 0,
             x, VD, (long long)VS * VD, VDH,
             VS, VDH, VS, VB, VNH, 1.f, GF_ADDC);

        // LN2
        layernorm_kernel<<<BS, 256, 0, stream>>>(x, ln2_g + (size_t)l * VD,
                                                 ln2_b + (size_t)l * VD, hln, VD);

        // MLP: m1 = gelu(hln @ W1^T + b1) ; x += m1 @ W2^T + b2
        gemm(hln, VD, 0, 0, W1 + (size_t)l * VMD * VD, VD, 0, 0,
             b1 + (size_t)l * VMD, 0, 0,
             m1, VMD, 0, 0, BS, VMD, VD, 1, 1, 1.f, GF_GELU);
        gemm(m1, VMD, 0, 0, W2 + (size_t)l * VD * VMD, VMD, 0, 0,
             b2 + (size_t)l * VD, 0, 0,
             x, VD, 0, 0, BS, VD, VMD, 1, 1, 1.f, GF_ADDC);
    }

    // ---- head: logits = x[:,0,:] @ Wh^T + bh ; softmax -> d_out ----
    gemm(x, VS * VD, 0, 0, Wh, VD, 0, 0, bh, 0, 0,
         logits, VOUT, 0, 0, VB, VOUT, VD, 1, 1, 1.f, 0);
    softmax_kernel<<<VB, 256, 0, stream>>>(logits, (float*)d_out, VOUT);
}